// Simple_AD_module_20882130993178
// MI455X (gfx1250) — compile-verified
//
#include <hip/hip_runtime.h>
#include <hip/hip_bf16.h>
#include <math.h>

#define T_STEPS 16
#define S_SAMP  8
#define DIM     8
#define NCELL   (T_STEPS * S_SAMP)   // 128 cells
#define MC      100000
#define EPSV    1e-6f
#define NBATCH  (MC / 16)            // 6250 batches of 16 samples
#define BPC     8                    // blocks per cell
#define WPB     8                    // waves per block (256 threads, wave32)
#define WPC     (BPC * WPB)          // 64 waves per cell
#define CELL_STRIDE 16               // floats per cell in workspace

typedef __attribute__((ext_vector_type(2))) float v2f;
typedef __attribute__((ext_vector_type(8))) float v8f;

// ---------------- stateless-seeded RNG helpers ----------------
__device__ __forceinline__ unsigned mix32(unsigned x) {
  x += 0x9E3779B9u; x ^= x >> 16; x *= 0x85EBCA6Bu;
  x ^= x >> 13;     x *= 0xC2B2AE35u; x ^= x >> 16;
  return x | 1u;
}
__device__ __forceinline__ float u01(unsigned& s) {
  s ^= s << 13; s ^= s >> 17; s ^= s << 5;
  return ((s >> 8) + 0.5f) * (1.0f / 16777216.0f);   // (0,1)
}
__device__ __forceinline__ float nrm(unsigned& s) {
  float u1 = u01(s), u2 = u01(s);
  return sqrtf(-2.0f * __logf(u1)) * __cosf(6.28318530718f * u2);
}
// Marsaglia-Tsang gamma(shape) with precomputed d = a'-1/3, c = 1/sqrt(9d);
// invA > 0 => boost path for shape < 1 (multiply by U^(1/a)).
__device__ __forceinline__ float gamma_mt(unsigned& s, float d, float c, float invA) {
  float g = d;  // fallback (also NaN-propagating when d is NaN)
  #pragma unroll 1
  for (int t = 0; t < 12; ++t) {
    float x = nrm(s);
    float v = 1.0f + c * x;
    if (v <= 0.0f) continue;
    v = v * v * v;
    float u  = u01(s);
    float x2 = x * x;
    if (u < 1.0f - 0.0331f * x2 * x2)                         { g = d * v; break; }
    if (__logf(u) < 0.5f * x2 + d - d * v + d * __logf(v))    { g = d * v; break; }
  }
  if (invA > 0.0f) g *= __powf(u01(s), invA);
  return g;
}

// ---------------- kernel 1: per-cell Dirichlet fit ----------------
__global__ void dirichlet_setup_kernel(const float* __restrict__ obs,
                                       const float* __restrict__ mom,
                                       float* __restrict__ wf,
                                       int* __restrict__ counts) {
  int cell = blockIdx.x * blockDim.x + threadIdx.x;
  if (cell >= NCELL) return;
  counts[cell] = 0;

  float e[DIM], v[DIM], o[DIM];
  #pragma unroll
  for (int d = 0; d < DIM; ++d) {
    e[d] = mom[(cell * DIM + d) * 2 + 0];
    v[d] = fmaxf(mom[(cell * DIM + d) * 2 + 1], 0.0f);
    o[d] = obs[cell * DIM + d];
  }
  // E = max(e, eps) / sum
  float E[DIM], se = 0.0f;
  #pragma unroll
  for (int d = 0; d < DIM; ++d) { E[d] = fmaxf(e[d], EPSV); se += E[d]; }
  #pragma unroll
  for (int d = 0; d < DIM; ++d) E[d] /= se;

  // factors = E(1-E)/v - 1; nanmedian over valid entries
  float f[DIM]; int n = 0;
  #pragma unroll
  for (int d = 0; d < DIM; ++d) {
    float fd = E[d] * (1.0f - E[d]) / v[d] - 1.0f;
    if (fd > 0.0f && v[d] > 0.0f) f[n++] = fd;
  }
  float factor;
  if (n == 0) {
    factor = __int_as_float(0x7fc00000);  // NaN -> propagates like reference
  } else {
    for (int i = 1; i < n; ++i) {         // insertion sort, n <= 8
      float key = f[i]; int j = i - 1;
      while (j >= 0 && f[j] > key) { f[j + 1] = f[j]; --j; }
      f[j + 1] = key;
    }
    factor = 0.5f * (f[(n - 1) / 2] + f[n / 2]);
  }

  float beta[DIM];  // beta = alpha - 1
  #pragma unroll
  for (int d = 0; d < DIM; ++d) beta[d] = E[d] * factor - 1.0f;

  // obs threshold: sum beta_d * log(norm_simplex(o)_d); gammaln terms cancel
  float oc[DIM], so = 0.0f;
  #pragma unroll
  for (int d = 0; d < DIM; ++d) { oc[d] = fmaxf(o[d], EPSV); so += oc[d]; }
  float lso = __logf(so), od = 0.0f;
  #pragma unroll
  for (int d = 0; d < DIM; ++d) od += beta[d] * (__logf(oc[d]) - lso);

  float* c = wf + cell * CELL_STRIDE;
  #pragma unroll
  for (int d = 0; d < DIM; ++d) c[d] = beta[d];
  c[8] = od;
}

// ---------------- kernel 2: Monte-Carlo with FP32 WMMA dot ----------------
__global__ void __launch_bounds__(256) dirichlet_mc_kernel(const float* __restrict__ wf,
                                                           int* __restrict__ counts) {
  const int cell = blockIdx.y;
  const int tid  = threadIdx.x;
  const int lane = tid & 31;
  const int half = lane >> 4;                // 0: dims {0,1,4,5}, 1: dims {2,3,6,7}
  // Wave-uniform by construction; assert via readfirstlane so the batch loop
  // compiles as a scalar (SGPR) loop: no EXEC churn, EXEC is structurally
  // all-1s at the WMMA sites (ISA requirement).
  const int wave_in_cell =
      __builtin_amdgcn_readfirstlane((int)(blockIdx.x * WPB) + (tid >> 5));

  const float* cw = wf + cell * CELL_STRIDE;
  const float obs_dot = cw[8];

  const int dm[4] = {2 * half, 2 * half + 1, 2 * half + 4, 2 * half + 5};
  float beta_l[4], gd[4], gc[4], ginvA[4];
  #pragma unroll
  for (int j = 0; j < 4; ++j) {
    float b = cw[dm[j]];
    beta_l[j] = b;
    float a = b + 1.0f, invA = 0.0f, ap = a;
    if (a < 1.0f) { ap = a + 1.0f; invA = 1.0f / a; }
    float dd = ap - (1.0f / 3.0f);
    gd[j] = dd;
    gc[j] = rsqrtf(9.0f * dd);
    ginvA[j] = invA;
  }
  // B operands of the 16x16x4 FP32 WMMA: (alpha-1) broadcast over all 16 columns.
  // K-slot layout matches this lane's dims exactly (K0/K2 in v0, K1/K3 in v1).
  v2f b1; b1.x = beta_l[0]; b1.y = beta_l[1];
  v2f b2; b2.x = beta_l[2]; b2.y = beta_l[3];

  unsigned seed = mix32((unsigned)(((cell * BPC + blockIdx.x) * 256 + tid)) * 0x9E3779B9u + 12345u);

  unsigned cnt = 0;  // wave-uniform via ballot accumulation
  for (int b = wave_in_cell; b < NBATCH; b += WPC) {
    // lane pair (L, L+16) jointly produces sample m = L%16 (4 gamma dims each)
    float g[4], psum = 0.0f;
    #pragma unroll
    for (int j = 0; j < 4; ++j) { g[j] = gamma_mt(seed, gd[j], gc[j], ginvA[j]); psum += g[j]; }
    float gsum = psum + __shfl_xor(psum, 16, 32);      // full D=8 gamma sum
    float y[4], pys = 0.0f;
    #pragma unroll
    for (int j = 0; j < 4; ++j) { y[j] = fmaxf(g[j] / gsum, EPSV); pys += y[j]; }
    float ysum = pys + __shfl_xor(pys, 16, 32);        // renorm after clip (norm_simplex)
    float lys = __logf(ysum);

    // A operands: log(z_d) laid out per the 16x4 FP32 A-matrix spec
    v2f a1, a2;
    a1.x = __logf(y[0]) - lys; a1.y = __logf(y[1]) - lys;   // dims 0..3 (K=0..3)
    a2.x = __logf(y[2]) - lys; a2.y = __logf(y[3]) - lys;   // dims 4..7

    v8f c = {};
    c = __builtin_amdgcn_wmma_f32_16x16x4_f32(false, a1, false, b1, (short)0, c, false, false);
    c = __builtin_amdgcn_wmma_f32_16x16x4_f32(false, a2, false, b2, (short)0, c, false, false);

    // Each sample's dot is replicated across 16 columns (x16 overcount, divided
    // out at the end). ballot+popcount keeps only the v_cmp on VALU; the
    // accumulation runs on SALU (s_bcnt/s_add), co-executing with TRANS math.
    #pragma unroll
    for (int r = 0; r < 8; ++r)
      cnt += (unsigned)__builtin_popcount(__builtin_amdgcn_ballot_w32(obs_dot > c[r]));
  }
  if (lane == 0) atomicAdd(&counts[cell], (int)cnt);
}

// ---------------- kernel 3: p-value -> sigmoid(-log(p+eps)), transpose ----------------
__global__ void dirichlet_final_kernel(const int* __restrict__ counts,
                                       const int* __restrict__ observed,
                                       float* __restrict__ out) {
  int cell = blockIdx.x * blockDim.x + threadIdx.x;
  if (cell >= NCELL) return;
  int t = cell / S_SAMP, s = cell % S_SAMP;
  float pval  = (float)counts[cell] * (1.0f / (16.0f * (float)MC)); // /16: column replication
  float score = -__logf(pval + EPSV);
  float r = 1.0f / (1.0f + __expf(-score));
  if (observed[cell] == 0) r = __int_as_float(0x7fc00000);
  out[s * T_STEPS + t] = r;   // [S, T] transposed layout
}

extern "C" void kernel_launch(void* const* d_in, const int* in_sizes, int n_in,
                              void* d_out, int out_size, void* d_ws, size_t ws_size,
                              hipStream_t stream) {
  (void)in_sizes; (void)n_in; (void)out_size; (void)ws_size;
  const float* obs      = (const float*)d_in[0];   // [T,S,D]
  const float* mom      = (const float*)d_in[1];   // [T,S,D,2]
  const int*   observed = (const int*)d_in[2];     // [T,S]
  float* wf     = (float*)d_ws;
  int*   counts = (int*)((char*)d_ws + NCELL * CELL_STRIDE * sizeof(float));

  dirichlet_setup_kernel<<<1, NCELL, 0, stream>>>(obs, mom, wf, counts);
  dirichlet_mc_kernel<<<dim3(BPC, NCELL), 256, 0, stream>>>(wf, counts);
  dirichlet_final_kernel<<<1, NCELL, 0, stream>>>(counts, observed, (float*)d_out);
}